// XYZConverter_5265629905635
// MI455X (gfx1250) — compile-verified
//
#include <hip/hip_runtime.h>
#include <math.h>

typedef float v2f __attribute__((ext_vector_type(2)));
typedef float v8f __attribute__((ext_vector_type(8)));

#define THREADS 256

// ---------------------------------------------------------------------------
// Wave-cooperative 4x4 matrix product via V_WMMA_F32_16X16X4_F32.
// 4 residues are batched block-diagonally per WMMA:
//   A (16x4)  = 4 left matrices stacked vertically
//   B (4x16)  = 4 right matrices side by side
//   C (16x16) diagonal 4x4 blocks = per-residue products.
// Staging through LDS; everything stays within one wave32 (wave-order LDS).
// ---------------------------------------------------------------------------
__device__ __forceinline__ void mat4mul_wave(const float* L, const float* R, float* out,
                                             float* Ls, float* Rs, int tid) {
  // stage L row-major; R column-major (so B lane fetches are contiguous pairs)
#pragma unroll
  for (int i = 0; i < 4; ++i)
    *(float4*)&Ls[tid * 16 + i * 4] =
        make_float4(L[i * 4 + 0], L[i * 4 + 1], L[i * 4 + 2], L[i * 4 + 3]);
#pragma unroll
  for (int j = 0; j < 4; ++j)
    *(float4*)&Rs[tid * 16 + j * 4] =
        make_float4(R[j], R[4 + j], R[8 + j], R[12 + j]);
  __builtin_amdgcn_wave_barrier();

  const int lane  = tid & 31;
  const int wbase = tid & ~31;          // this wave's 32 residues in the stage
  const int nloc  = lane & 15;          // row (A) / column (B) index 0..15
  const int kp    = (lane >> 4) << 1;   // K pair: lanes 0-15 -> K0,K1 ; 16-31 -> K2,K3
  const int blk   = nloc >> 2;          // which 4x4 block within the 16
  const int jcol  = nloc & 3;
  const bool hi   = (blk & 1) != 0;     // M-block 1/3 -> accumulator regs 4..7
  // C diag elements live in lanes 0-7 (M=0..7) and 24-31 (M=8..15)
  const bool writer = (((lane >> 3) & 1) == (lane >> 4));

#pragma unroll
  for (int g = 0; g < 8; ++g) {         // 8 groups of 4 residues = 32 products
    const int rb  = wbase + g * 4;
    const int off = (rb + blk) * 16 + jcol * 4 + kp;  // same formula for A and B
    v2f a = *(const v2f*)&Ls[off];      // A(m, kp..kp+1), m = nloc
    v2f b = *(const v2f*)&Rs[off];      // B(kp..kp+1, n), n = nloc (R col-major)
    v8f c = {0.f, 0.f, 0.f, 0.f, 0.f, 0.f, 0.f, 0.f};
    c = __builtin_amdgcn_wmma_f32_16x16x4_f32(false, a, false, b,
                                              (short)0, c, false, false);
    if (writer) {
      // constant-index extracts first, then scalar selects -> 4 v_cndmask
      float e0 = c[0], e1 = c[1], e2 = c[2], e3 = c[3];
      float e4 = c[4], e5 = c[5], e6 = c[6], e7 = c[7];
      float4 col = make_float4(hi ? e4 : e0, hi ? e5 : e1,
                               hi ? e6 : e2, hi ? e7 : e3);
      *(float4*)&Ls[(rb + blk) * 16 + jcol * 4] = col;  // result, column-major
    }
    __builtin_amdgcn_wave_barrier();
  }
  // read back own 4x4 (column-major in Ls region)
#pragma unroll
  for (int j = 0; j < 4; ++j) {
    float4 cj = *(float4*)&Ls[tid * 16 + j * 4];
    out[0 * 4 + j] = cj.x; out[1 * 4 + j] = cj.y;
    out[2 * 4 + j] = cj.z; out[3 * 4 + j] = cj.w;
  }
}

// M <- M * rotX(a)   (only columns 1,2 change)
__device__ __forceinline__ void apply_rotX(float* M, float a0, float a1) {
  float n = sqrtf(a0 * a0 + a1 * a1) + 1e-6f;
  float c = a0 / n, s = a1 / n;
#pragma unroll
  for (int i = 0; i < 4; ++i) {
    float m1 = M[i * 4 + 1], m2 = M[i * 4 + 2];
    M[i * 4 + 1] =  c * m1 + s * m2;
    M[i * 4 + 2] = -s * m1 + c * m2;
  }
}
// M <- M * rotZ(a)   (only columns 0,1 change)
__device__ __forceinline__ void apply_rotZ(float* M, float a0, float a1) {
  float n = sqrtf(a0 * a0 + a1 * a1) + 1e-6f;
  float c = a0 / n, s = a1 / n;
#pragma unroll
  for (int i = 0; i < 4; ++i) {
    float m0 = M[i * 4 + 0], m1 = M[i * 4 + 1];
    M[i * 4 + 0] =  c * m0 + s * m1;
    M[i * 4 + 1] = -s * m0 + c * m1;
  }
}

__device__ __forceinline__ void make_rot_axis_m(float* R, float a0, float a1,
                                                float u0, float u1, float u2) {
  float n = sqrtf(a0 * a0 + a1 * a1) + 1e-6f;
  float ct = a0 / n, st = a1 / n, omc = 1.0f - ct;
  R[0] = ct + u0 * u0 * omc;      R[1] = u0 * u1 * omc - u2 * st;
  R[2] = u0 * u2 * omc + u1 * st; R[3] = 0.f;
  R[4] = u0 * u1 * omc + u2 * st; R[5] = ct + u1 * u1 * omc;
  R[6] = u1 * u2 * omc - u0 * st; R[7] = 0.f;
  R[8] = u0 * u2 * omc - u1 * st; R[9] = u1 * u2 * omc + u0 * st;
  R[10] = ct + u2 * u2 * omc;     R[11] = 0.f;
  R[12] = 0.f; R[13] = 0.f; R[14] = 0.f; R[15] = 1.f;
}

__device__ __forceinline__ void load16(float* M, const float* g) {
#pragma unroll
  for (int i = 0; i < 4; ++i) {
    float4 v = *(const float4*)(g + i * 4);
    M[i * 4] = v.x; M[i * 4 + 1] = v.y; M[i * 4 + 2] = v.z; M[i * 4 + 3] = v.w;
  }
}
__device__ __forceinline__ void store16(float* g, const float* M) {
#pragma unroll
  for (int i = 0; i < 4; ++i)
    *(float4*)(g + i * 4) =
        make_float4(M[i * 4], M[i * 4 + 1], M[i * 4 + 2], M[i * 4 + 3]);
}

// ---------------------------------------------------------------------------
// Kernel 1: one residue per thread; 17 frames via wave-coop WMMA chain.
// Grid is exact (B*L divisible by 256) so EXEC is all-ones at every WMMA.
// ---------------------------------------------------------------------------
__global__ __launch_bounds__(THREADS)
void xyzconv_frames_kernel(const float* __restrict__ xyz, const float* __restrict__ alphas,
                           const float* __restrict__ RTs, const float* __restrict__ xyzs_base,
                           const int* __restrict__ seq, float* __restrict__ RTout) {
  __shared__ __align__(16) float Ls[THREADS * 16];
  __shared__ __align__(16) float Rs[THREADS * 16];
  const int tid = threadIdx.x;
  const int r = blockIdx.x * THREADS + tid;

  // ---- rigid_from_3_points -> RTF0 ----
  const float* x = xyz + (size_t)r * 9;
  float Nx = x[0], Ny = x[1], Nz = x[2];
  float Ax = x[3], Ay = x[4], Az = x[5];
  float Cx = x[6], Cy = x[7], Cz = x[8];
  float v1x = Cx - Ax, v1y = Cy - Ay, v1z = Cz - Az;
  float v2x = Nx - Ax, v2y = Ny - Ay, v2z = Nz - Az;
  float i1 = 1.0f / (sqrtf(v1x * v1x + v1y * v1y + v1z * v1z) + 1e-8f);
  float e1x = v1x * i1, e1y = v1y * i1, e1z = v1z * i1;
  float d = e1x * v2x + e1y * v2y + e1z * v2z;
  float u2x = v2x - d * e1x, u2y = v2y - d * e1y, u2z = v2z - d * e1z;
  float i2 = 1.0f / (sqrtf(u2x * u2x + u2y * u2y + u2z * u2z) + 1e-8f);
  float e2x = u2x * i2, e2y = u2y * i2, e2z = u2z * i2;
  float e3x = e1y * e2z - e1z * e2y;
  float e3y = e1z * e2x - e1x * e2z;
  float e3z = e1x * e2y - e1y * e2x;

  float F0[16] = { e1x, e2x, e3x, Ax,  e1y, e2y, e3y, Ay,
                   e1z, e2z, e3z, Az,  0.f, 0.f, 0.f, 1.f };

  const int s = seq[r];
  const float* bRT = RTs + (size_t)s * 17 * 16;
  const float* al  = alphas + (size_t)r * 40;
  float* fr = RTout + (size_t)r * 272;

  store16(fr + 0 * 16, F0);

  float T[16], P[16], Q[16];

  // F1..F3 = F0 * bRT[k] * rotX(k)
  load16(T, bRT + 0 * 16); mat4mul_wave(F0, T, P, Ls, Rs, tid); apply_rotX(P, al[0], al[1]); store16(fr + 1 * 16, P);
  load16(T, bRT + 1 * 16); mat4mul_wave(F0, T, P, Ls, Rs, tid); apply_rotX(P, al[2], al[3]); store16(fr + 2 * 16, P);
  load16(T, bRT + 2 * 16); mat4mul_wave(F0, T, P, Ls, Rs, tid); apply_rotX(P, al[4], al[5]); store16(fr + 3 * 16, P);

  // ---- CB rotation axes from template coordinates ----
  const float* bx = xyzs_base + (size_t)s * 144;
  float n0x = bx[0],  n0y = bx[1],  n0z = bx[2];
  float cax = bx[4],  cay = bx[5],  caz = bx[6];
  float c2x = bx[8],  c2y = bx[9],  c2z = bx[10];
  float cbx = bx[16], cby = bx[17], cbz = bx[18];
  float ncrx = 0.5f * (c2x + n0x), ncry = 0.5f * (c2y + n0y), ncrz = 0.5f * (c2z + n0z);
  float vax = cbx - cax, vay = cby - cay, vaz = cbz - caz;
  float vbx = ncrx - cax, vby = ncry - cay, vbz = ncrz - caz;
  float a1x = vay * vbz - vaz * vby, a1y = vaz * vbx - vax * vbz, a1z = vax * vby - vay * vbx;
  float ia1 = 1.0f / (sqrtf(a1x * a1x + a1y * a1y + a1z * a1z) + 1e-8f);
  a1x *= ia1; a1y *= ia1; a1z *= ia1;
  float npx = c2x - n0x, npy = c2y - n0y, npz = c2z - n0z;
  float tq = (npx * ncrx + npy * ncry + npz * ncrz) /
             (ncrx * ncrx + ncry * ncry + ncrz * ncrz);
  float nppx = npx - tq * ncrx, nppy = npy - tq * ncry, nppz = npz - tq * ncrz;
  float a2x = vay * nppz - vaz * nppy, a2y = vaz * nppx - vax * nppz, a2z = vax * nppy - vay * nppx;
  float ia2 = 1.0f / (sqrtf(a2x * a2x + a2y * a2y + a2z * a2z) + 1e-8f);
  a2x *= ia2; a2y *= ia2; a2z *= ia2;

  // F8 = F0 * CBrot1 * CBrot2
  make_rot_axis_m(T, al[14], al[15], a1x, a1y, a1z);
  mat4mul_wave(F0, T, P, Ls, Rs, tid);
  make_rot_axis_m(T, al[16], al[17], a2x, a2y, a2z);
  mat4mul_wave(P, T, Q, Ls, Rs, tid);
  store16(fr + 8 * 16, Q);

  // F4 = F8*bRT3*rX3*rZ9 ; F5..F7 chain
  load16(T, bRT + 3 * 16); mat4mul_wave(Q, T, P, Ls, Rs, tid);
  apply_rotX(P, al[6], al[7]); apply_rotZ(P, al[18], al[19]); store16(fr + 4 * 16, P);
  load16(T, bRT + 4 * 16); mat4mul_wave(P, T, Q, Ls, Rs, tid); apply_rotX(Q, al[8],  al[9]);  store16(fr + 5 * 16, Q);
  load16(T, bRT + 5 * 16); mat4mul_wave(Q, T, P, Ls, Rs, tid); apply_rotX(P, al[10], al[11]); store16(fr + 6 * 16, P);
  load16(T, bRT + 6 * 16); mat4mul_wave(P, T, Q, Ls, Rs, tid); apply_rotX(Q, al[12], al[13]); store16(fr + 7 * 16, Q);

  // F9..F16 chain (F11 and F14 each feed two children)
  load16(T, bRT + 9 * 16);  mat4mul_wave(F0, T, P, Ls, Rs, tid); apply_rotX(P, al[24], al[25]); store16(fr + 9 * 16, P);
  load16(T, bRT + 10 * 16); mat4mul_wave(P, T, Q, Ls, Rs, tid);  apply_rotX(Q, al[26], al[27]); store16(fr + 10 * 16, Q);
  load16(T, bRT + 11 * 16); mat4mul_wave(Q, T, P, Ls, Rs, tid);  apply_rotX(P, al[28], al[29]); store16(fr + 11 * 16, P); // P=F11
  load16(T, bRT + 12 * 16); mat4mul_wave(P, T, Q, Ls, Rs, tid);  apply_rotX(Q, al[30], al[31]); store16(fr + 12 * 16, Q);
  load16(T, bRT + 13 * 16); mat4mul_wave(P, T, Q, Ls, Rs, tid);  apply_rotX(Q, al[32], al[33]); store16(fr + 13 * 16, Q); // Q=F13
  load16(T, bRT + 14 * 16); mat4mul_wave(Q, T, P, Ls, Rs, tid);  apply_rotX(P, al[34], al[35]); store16(fr + 14 * 16, P); // P=F14
  load16(T, bRT + 15 * 16); mat4mul_wave(P, T, Q, Ls, Rs, tid);  apply_rotX(Q, al[36], al[37]); store16(fr + 15 * 16, Q);
  load16(T, bRT + 16 * 16); mat4mul_wave(P, T, Q, Ls, Rs, tid);  apply_rotX(Q, al[38], al[39]); store16(fr + 16 * 16, Q);
}

// ---------------------------------------------------------------------------
// Kernel 2: one atom per thread; gathers frames (L2-resident: 142MB < 192MB L2)
// ---------------------------------------------------------------------------
__global__ __launch_bounds__(THREADS)
void xyzconv_atoms_kernel(const float* __restrict__ RTframes, const float* __restrict__ xyzs_base,
                          const int* __restrict__ seq, const int* __restrict__ base_idx,
                          float* __restrict__ xyzout, int total) {
  int gid = blockIdx.x * THREADS + threadIdx.x;
  if (gid >= total) return;
  int r = gid / 36;
  int t = gid - r * 36;
  int s = seq[r];
  int idx = base_idx[s * 36 + t];
  const float* M = RTframes + (size_t)r * 272 + (size_t)idx * 16;
  float4 m0 = *(const float4*)(M + 0);
  float4 m1 = *(const float4*)(M + 4);
  float4 m2 = *(const float4*)(M + 8);
  float4 p  = *(const float4*)(xyzs_base + ((size_t)s * 36 + t) * 4);
  float o0 = m0.x * p.x + m0.y * p.y + m0.z * p.z + m0.w * p.w;
  float o1 = m1.x * p.x + m1.y * p.y + m1.z * p.z + m1.w * p.w;
  float o2 = m2.x * p.x + m2.y * p.y + m2.z * p.z + m2.w * p.w;
  float* o = xyzout + (size_t)gid * 3;
  o[0] = o0; o[1] = o1; o[2] = o2;
}

extern "C" void kernel_launch(void* const* d_in, const int* in_sizes, int n_in,
                              void* d_out, int out_size, void* d_ws, size_t ws_size,
                              hipStream_t stream) {
  const float* xyz    = (const float*)d_in[0];
  const float* alphas = (const float*)d_in[1];
  const float* RTs    = (const float*)d_in[2];
  const float* xb     = (const float*)d_in[3];
  const int*   seq    = (const int*)d_in[4];
  const int*   bidx   = (const int*)d_in[5];

  const int nres = in_sizes[4];                 // B*L = 131072 (divisible by 256)
  float* rtf  = (float*)d_out;                  // RTframes: nres*17*16 floats
  float* xyzo = rtf + (size_t)nres * 272;       // xyzs:     nres*36*3 floats

  int blocksA = nres / THREADS;
  xyzconv_frames_kernel<<<blocksA, THREADS, 0, stream>>>(xyz, alphas, RTs, xb, seq, rtf);

  int total = nres * 36;
  int blocksB = (total + THREADS - 1) / THREADS;
  xyzconv_atoms_kernel<<<blocksB, THREADS, 0, stream>>>(rtf, xb, seq, bidx, xyzo, total);
}